// ScatterEdges_77790447665656
// MI455X (gfx1250) — compile-verified
//
#include <hip/hip_runtime.h>

// ScatterEdges for MI455X (gfx1250, wave32).
//   out[src[e]] += x[e]*switch[e];  out[dst[e]] -= x[e]*switch[e]
// HBM-bound: ~330 MB traffic -> ~14 us floor @ 23.3 TB/s. Output (9.6 MB)
// stays resident in the 192 MB L2; all scatter atomics resolve there.
// x is streamed through LDS with gfx1250 async loads
// (global_load_async_to_lds_b128, ASYNCcnt, th:TH_LOAD_NT) so the read-once
// stream never evicts the output working set from L2.

typedef float v4f __attribute__((ext_vector_type(4)));

#define DFEAT       48
#define TILE_E      128                      // edges per block tile
#define TILE_FLOATS (TILE_E * DFEAT)         // 6144 floats (24 KB LDS)
#define TILE_CHUNKS (TILE_FLOATS / 4)        // 1536 float4 chunks
#define BLOCK       256
#define CHUNK_ITERS (TILE_CHUNKS / BLOCK)    // 6

__device__ __forceinline__ void async_tile_b128(unsigned lds_addr, unsigned off,
                                                const char* gbase) {
    asm volatile("global_load_async_to_lds_b128 %0, %1, %2 th:TH_LOAD_NT"
                 :
                 : "v"(lds_addr), "v"(off), "s"(gbase)
                 : "memory");
}

__device__ __forceinline__ void scatter_chunk(const float* __restrict__ tile,
                                              int chunk, int e0,
                                              const float* __restrict__ sw,
                                              const int* __restrict__ esrc,
                                              const int* __restrict__ edst,
                                              float* __restrict__ out) {
    const int el = chunk / 12;                  // edge within tile (magic mul)
    const int c  = chunk - el * 12;             // float4 within the 48-f row

    v4f xv = *(const v4f*)&tile[chunk * 4];     // ds_load_b128
    const int e = e0 + el;
    const float s = sw[e];                      // 12 threads share -> L0 hit
    v4f w = xv * s;

    const int bs = esrc[e] * DFEAT + c * 4;
    const int bd = edst[e] * DFEAT + c * 4;

    // No-return hw f32 atomics -> L2 atomic units (output L2-resident).
#pragma unroll
    for (int k = 0; k < 4; ++k) {
        unsafeAtomicAdd(out + bs + k,  w[k]);
        unsafeAtomicAdd(out + bd + k, -w[k]);
    }
}

__global__ __launch_bounds__(BLOCK) void ScatterEdges_scatter_kernel(
    const float* __restrict__ x,     // [E, 48]
    const float* __restrict__ sw,    // [E]
    const int*  __restrict__ esrc,   // [E]
    const int*  __restrict__ edst,   // [E]
    float* __restrict__ out,         // [N, 48]
    int n_edges)
{
    __shared__ __align__(16) float tile[TILE_FLOATS];   // 24 KB of 320 KB/WGP

    const int tid = threadIdx.x;
    const int e0  = blockIdx.x * TILE_E;
    int tile_e    = n_edges - e0;
    if (tile_e > TILE_E) tile_e = TILE_E;

    // Uniform 64-bit global base of this block's x tile (SGPR pair in asm).
    const char* gbase = (const char*)(x + (size_t)e0 * DFEAT);
    // Low 32 bits of the generic shared address == LDS byte offset.
    const unsigned lds_base = (unsigned)(uintptr_t)(void*)tile;

    if (tile_e == TILE_E) {
        // ---- Fast path: full tile (all blocks when E % TILE_E == 0) ----
        // 6 async b128s issued back-to-back, no exec-mask juggling.
#pragma unroll
        for (int j = 0; j < CHUNK_ITERS; ++j) {
            const unsigned off = (unsigned)(j * BLOCK + tid) * 16u;
            async_tile_b128(lds_base + off, off, gbase);
        }
        asm volatile("s_wait_asynccnt 0x0" ::: "memory");
        __syncthreads();

#pragma unroll
        for (int j = 0; j < CHUNK_ITERS; ++j) {
            scatter_chunk(tile, j * BLOCK + tid, e0, sw, esrc, edst, out);
        }
    } else {
        // ---- Tail path: ragged last tile (at most one block) ----
        const int valid_bytes = tile_e * DFEAT * 4;
#pragma unroll
        for (int j = 0; j < CHUNK_ITERS; ++j) {
            const unsigned off = (unsigned)(j * BLOCK + tid) * 16u;
            if ((int)off < valid_bytes) {
                async_tile_b128(lds_base + off, off, gbase);
            }
        }
        asm volatile("s_wait_asynccnt 0x0" ::: "memory");
        __syncthreads();

#pragma unroll
        for (int j = 0; j < CHUNK_ITERS; ++j) {
            const int chunk = j * BLOCK + tid;
            if (chunk / 12 < tile_e) {
                scatter_chunk(tile, chunk, e0, sw, esrc, edst, out);
            }
        }
    }
}

__global__ __launch_bounds__(BLOCK) void ScatterEdges_zero_kernel(
    float* __restrict__ out, int n4, int n)
{
    int i = blockIdx.x * blockDim.x + threadIdx.x;
    if (i < n4) {
        ((v4f*)out)[i] = (v4f)(0.0f);
    }
    if (i == 0) {                 // scalar tail if n % 4 != 0
        for (int k = n4 * 4; k < n; ++k) out[k] = 0.0f;
    }
}

extern "C" void kernel_launch(void* const* d_in, const int* in_sizes, int n_in,
                              void* d_out, int out_size, void* d_ws, size_t ws_size,
                              hipStream_t stream) {
    const float* x    = (const float*)d_in[0];  // [E, 48] f32
    const float* sw   = (const float*)d_in[1];  // [E] f32
    const int*   esrc = (const int*)d_in[2];    // [E] i32
    const int*   edst = (const int*)d_in[3];    // [E] i32
    // d_in[4] = species: only defines N; not read on device.
    float* out = (float*)d_out;

    const int n_edges = in_sizes[1];            // E from switch
    const int n4      = out_size / 4;

    // 1) zero the (poisoned) output
    {
        int blocks = (n4 + BLOCK - 1) / BLOCK;
        if (blocks < 1) blocks = 1;
        ScatterEdges_zero_kernel<<<blocks, BLOCK, 0, stream>>>(out, n4, out_size);
    }

    // 2) async-staged scale + antisymmetric scatter
    {
        int blocks = (n_edges + TILE_E - 1) / TILE_E;   // 12500 tiles
        ScatterEdges_scatter_kernel<<<blocks, BLOCK, 0, stream>>>(
            x, sw, esrc, edst, out, n_edges);
    }
}